// OCWM_9285719294560
// MI455X (gfx1250) — compile-verified
//
#include <hip/hip_runtime.h>
#include <hip/hip_bf16.h>

// Problem constants (must match reference)
#define Bn   256
#define Nn   1024
#define Kn   128
#define Dn   389            // 64 + 128 + 4 + 1 + 64 + 64 + 64
#define Tthr 0.5f
#define BK   (Bn * Kn)          // 32768
#define OBJ_OFF (BK * Dn)       // 12746752  (floats)
#define ID_OFF  (OBJ_OFF + Bn)  // 12747008  (floats)

// ---------------------------------------------------------------------------
// CDNA5 async Global<->LDS path (gfx1250). Builtin signatures (verified via
// hipcc diagnostics):
//   b32 : (int  AS1* gaddr, int  AS3* ldsaddr, imm int off, imm cpol)
//   b128: (vi4  AS1* gaddr, vi4  AS3* ldsaddr, imm int off, imm cpol)
//         where vi4 = int __attribute__((vector_size(16)))
// ---------------------------------------------------------------------------
#if defined(__gfx1250__) && \
    __has_builtin(__builtin_amdgcn_global_load_async_to_lds_b32) && \
    __has_builtin(__builtin_amdgcn_global_store_async_from_lds_b32) && \
    __has_builtin(__builtin_amdgcn_s_wait_asynccnt)
#define USE_ASYNC_LDS 1
#else
#define USE_ASYNC_LDS 0
#endif

#if USE_ASYNC_LDS && __has_builtin(__builtin_amdgcn_global_load_async_to_lds_b128)
#define HAVE_ASYNC_B128 1
#else
#define HAVE_ASYNC_B128 0
#endif

#if USE_ASYNC_LDS
typedef __attribute__((address_space(1))) int GInt;   // global (AS1) int*
typedef __attribute__((address_space(3))) int LInt;   // LDS    (AS3) int*

__device__ __forceinline__ void async_g2l_b32(const float* g, float* l) {
    __builtin_amdgcn_global_load_async_to_lds_b32((GInt*)g, (LInt*)l, 0, 0);
}
__device__ __forceinline__ void async_l2g_b32(float* g, const float* l) {
    __builtin_amdgcn_global_store_async_from_lds_b32((GInt*)g, (LInt*)l, 0, 0);
}
__device__ __forceinline__ void wait_async0() {
    __builtin_amdgcn_s_wait_asynccnt(0);
}
#endif

#if HAVE_ASYNC_B128
typedef int vi4 __attribute__((vector_size(4 * sizeof(int))));
typedef __attribute__((address_space(1))) vi4 GInt4;  // global (AS1) int4*
typedef __attribute__((address_space(3))) vi4 LInt4;  // LDS    (AS3) int4*

__device__ __forceinline__ void async_g2l_b128(const float* g, float* l) {
    __builtin_amdgcn_global_load_async_to_lds_b128((GInt4*)g, (LInt4*)l, 0, 0);
}
#endif

// ---------------------------------------------------------------------------
// Kernel 1: per-batch stable-argsort rank -> top-K indices + kept count.
// One 1024-thread block per batch. rank(i) = #{v_j < v_i} + #{v_j==v_i, j<i}.
// ---------------------------------------------------------------------------
__global__ __launch_bounds__(1024)
void topk_rank_kernel(const float* __restrict__ z_pres,
                      int* __restrict__ top_idx,     // [B*K]
                      int* __restrict__ obj_cnt) {   // [B]
    __shared__ __align__(16) float vals[Nn];
    __shared__ int s_cnt;
    const int b = blockIdx.x;
    const int i = threadIdx.x;

    const float v = z_pres[b * Nn + i];
    vals[i] = v;
    if (i == 0) s_cnt = 0;
    __syncthreads();

    int rank = 0;
#pragma unroll 4
    for (int j = 0; j < Nn; j += 4) {
        const float4 w4 = *reinterpret_cast<const float4*>(&vals[j]);
        rank += (w4.x < v) || (w4.x == v && (j + 0) < i);
        rank += (w4.y < v) || (w4.y == v && (j + 1) < i);
        rank += (w4.z < v) || (w4.z == v && (j + 2) < i);
        rank += (w4.w < v) || (w4.w == v && (j + 3) < i);
    }

    if (rank >= Nn - Kn) {
        const int k = rank - (Nn - Kn);          // slot 0..K-1 (ascending value)
        top_idx[b * Kn + k] = i;
        if (v > Tthr) atomicAdd(&s_cnt, 1);
    }
    __syncthreads();
    if (i == 0) obj_cnt[b] = s_cnt;
}

// ---------------------------------------------------------------------------
// Kernel 2: exclusive scan of obj_cnt over 256 batches -> batch_off[257];
// also emit obj_nums output (value-converted to float in concat buffer).
// ---------------------------------------------------------------------------
__global__ __launch_bounds__(Bn)
void prefix_kernel(const int* __restrict__ obj_cnt,
                   int* __restrict__ batch_off,     // [B+1]
                   float* __restrict__ out) {       // d_out base
    __shared__ int s[Bn];
    const int t = threadIdx.x;
    const int c = obj_cnt[t];
    s[t] = c;
    __syncthreads();
#pragma unroll
    for (int off = 1; off < Bn; off <<= 1) {
        int add = (t >= off) ? s[t - off] : 0;
        __syncthreads();
        s[t] += add;
        __syncthreads();
    }
    batch_off[t + 1] = s[t];                 // inclusive -> [1..B]
    if (t == 0) batch_off[0] = 0;
    out[OBJ_OFF + t] = (float)c;             // obj_nums
}

// ---------------------------------------------------------------------------
// Flat feature index f in [0,389) -> source element pointer.
// Concat order: z_what(64) z_what_h(128) z_where(4) z_pres(1)
//               z_where_h(64) z_pres_h(64) prior_h(64)
// ---------------------------------------------------------------------------
__device__ __forceinline__ const float*
feat_ptr(int f, int rowbase,
         const float* zw, const float* zwh, const float* zwr, const float* zp,
         const float* zrh, const float* zph, const float* prh) {
    if (f < 64)   return zw  + rowbase * 64  + f;
    if (f < 192)  return zwh + rowbase * 128 + (f - 64);
    if (f < 196)  return zwr + rowbase * 4   + (f - 192);
    if (f == 196) return zp  + rowbase;
    if (f < 261)  return zrh + rowbase * 64  + (f - 197);
    if (f < 325)  return zph + rowbase * 64  + (f - 261);
    return prh + rowbase * 64 + (f - 325);
}

// ---------------------------------------------------------------------------
// Kernel 3: one wave32 per (b,k) row. Kept rows (suffix k >= K-cnt):
//   gather:  4x per-lane async b128 loads (97 16B chunks) + 1 b32 (z_pres)
//            into a 16B-aligned padded LDS staging row,
//   s_wait_asynccnt 0,
//   scatter: 13x per-lane async b32 stores to the 4B-aligned compact row.
// Staged layout (floats): zw@0 zwh@64 zwr@192 zp@196 [pad 197..199]
//                         zrh@200 zph@264 prh@328  (all segment starts 16B)
// Logical f -> staged index: f < 197 ? f : f+3.
// ---------------------------------------------------------------------------
#define WAVES_PER_BLOCK 8
__global__ __launch_bounds__(32 * WAVES_PER_BLOCK)
void compact_kernel(const float* __restrict__ zp,  const float* __restrict__ zw,
                    const float* __restrict__ zwh, const float* __restrict__ zwr,
                    const float* __restrict__ zrh, const float* __restrict__ zph,
                    const float* __restrict__ prh, const int* __restrict__ obj_id,
                    const int* __restrict__ top_idx,
                    const int* __restrict__ batch_off,
                    float* __restrict__ out) {
    __shared__ __align__(16) float stage[WAVES_PER_BLOCK][400];

    const int lane = threadIdx.x & 31;
    const int w    = threadIdx.x >> 5;
    const int row  = blockIdx.x * WAVES_PER_BLOCK + w;   // [0, B*K)
    const int b    = row / Kn;
    const int k    = row % Kn;

    const int off0 = batch_off[b];
    const int cnt  = batch_off[b + 1] - off0;
    if (k < Kn - cnt) return;                 // not kept (keep-mask is a suffix)

    const int dest    = off0 + (k - (Kn - cnt));
    const int src     = top_idx[row];         // original object index n
    const int rowbase = b * Nn + src;
    float* gdst = out + dest * Dn;

#if USE_ASYNC_LDS
    float* const base = &stage[w][0];
#if HAVE_ASYNC_B128
    // 97 16-byte chunks: zw 0..15 | zwh 16..47 | zwr 48 | zrh 49..64
    //                    zph 65..80 | prh 81..96
#pragma unroll
    for (int it = 0; it < 4; ++it) {
        const int c = it * 32 + lane;
        if (c < 97) {
            const float* g;
            int l;                                     // staged float index
            if (c < 16)      { g = zw  + rowbase * 64  + c * 4;        l = c * 4;              }
            else if (c < 48) { g = zwh + rowbase * 128 + (c - 16) * 4; l = 64 + (c - 16) * 4;  }
            else if (c == 48){ g = zwr + rowbase * 4;                  l = 192;                }
            else if (c < 65) { g = zrh + rowbase * 64  + (c - 49) * 4; l = 200 + (c - 49) * 4; }
            else if (c < 81) { g = zph + rowbase * 64  + (c - 65) * 4; l = 264 + (c - 65) * 4; }
            else             { g = prh + rowbase * 64  + (c - 81) * 4; l = 328 + (c - 81) * 4; }
            async_g2l_b128(g, base + l);
        }
    }
    if (lane == 0) async_g2l_b32(zp + rowbase, base + 196);   // z_pres scalar
#else
    // fallback gather: 13 per-lane b32 loads into padded layout
#pragma unroll
    for (int it = 0; it < 13; ++it) {
        const int f = it * 32 + lane;
        if (f < Dn)
            async_g2l_b32(feat_ptr(f, rowbase, zw, zwh, zwr, zp, zrh, zph, prh),
                          base + (f < 197 ? f : f + 3));
    }
#endif
    wait_async0();                            // all gathers complete (in-order)
    // stream staged row -> global compact row (4B-aligned dest, stride 1556B)
#pragma unroll
    for (int it = 0; it < 13; ++it) {
        const int f = it * 32 + lane;
        if (f < Dn) async_l2g_b32(gdst + f, base + (f < 197 ? f : f + 3));
    }
#else
    (void)stage;
#pragma unroll
    for (int it = 0; it < 13; ++it) {
        const int f = it * 32 + lane;
        if (f < Dn)
            gdst[f] = *feat_ptr(f, rowbase, zw, zwh, zwr, zp, zrh, zph, prh);
    }
#endif

    if (lane == 0) out[ID_OFF + dest] = (float)obj_id[rowbase];
}

// ---------------------------------------------------------------------------
// Kernel 4: zero the contiguous tail [total, B*K) of compact + id buffers
// (d_out is poisoned by the harness; unkept rows must read as zero).
// ---------------------------------------------------------------------------
__global__ __launch_bounds__(256)
void zero_tail_kernel(const int* __restrict__ batch_off,
                      float* __restrict__ out) {
    const int total  = batch_off[Bn];         // total kept rows
    const int thrF   = total * Dn;            // first float of compact tail
    const int stride = gridDim.x * blockDim.x;
    const int tid    = blockIdx.x * blockDim.x + threadIdx.x;

    for (int idx = tid; idx < BK * Dn; idx += stride)
        if (idx >= thrF) out[idx] = 0.0f;

    for (int idx = tid; idx < BK; idx += stride)
        if (idx >= total) out[ID_OFF + idx] = 0.0f;
}

// ---------------------------------------------------------------------------
extern "C" void kernel_launch(void* const* d_in, const int* in_sizes, int n_in,
                              void* d_out, int out_size, void* d_ws, size_t ws_size,
                              hipStream_t stream) {
    // setup_inputs() order:
    const float* z_pres    = (const float*)d_in[0];  // [B,N]
    const float* z_what    = (const float*)d_in[1];  // [B,N,64]
    const float* z_what_h  = (const float*)d_in[2];  // [B,N,128]
    const float* z_where   = (const float*)d_in[3];  // [B,N,4]
    const float* z_where_h = (const float*)d_in[4];  // [B,N,64]
    const float* z_pres_h  = (const float*)d_in[5];  // [B,N,64]
    const float* prior_h   = (const float*)d_in[6];  // [B,N,64]
    const int*   obj_id    = (const int*)  d_in[7];  // [B,N]

    float* out = (float*)d_out;

    // workspace layout
    int* top_idx   = (int*)d_ws;                 // B*K ints
    int* obj_cnt   = top_idx + BK;               // B ints
    int* batch_off = obj_cnt + Bn;               // B+1 ints

    topk_rank_kernel<<<Bn, Nn, 0, stream>>>(z_pres, top_idx, obj_cnt);
    prefix_kernel<<<1, Bn, 0, stream>>>(obj_cnt, batch_off, out);
    compact_kernel<<<BK / WAVES_PER_BLOCK, 32 * WAVES_PER_BLOCK, 0, stream>>>(
        z_pres, z_what, z_what_h, z_where, z_where_h, z_pres_h, prior_h,
        obj_id, top_idx, batch_off, out);
    zero_tail_kernel<<<2048, 256, 0, stream>>>(batch_off, out);
}